// NormActiveAdaptDropWithLoss_3891240370806
// MI455X (gfx1250) — compile-verified
//
#include <hip/hip_runtime.h>
#include <stdint.h>

// ---------------------------------------------------------------------------
// Per-row k-smallest masking (adaptive dropout) for [16384 x 4096] fp32.
// One 256-thread block handles ROWS_PB rows, streaming each 16KB row into a
// double-buffered LDS staging area with gfx1250 async global->LDS loads
// (ASYNCcnt + s_wait_asynccnt, TH_LOAD_NT streaming hint), then running a
// 3-pass (11/11/10-bit) radix-select on order-preserving u32 keys held in
// registers. Block scans are wave32-native: in-wave shuffle scan + 8-entry
// cross-wave combine (2 barriers per scan). Output uses non-temporal b128
// stores (write-once stream; keep the 192MB L2 clean).
// ---------------------------------------------------------------------------

#define FEAT      4096
#define THREADS   256
#define NWAVES    (THREADS / 32)
#define PER_TH    (FEAT / THREADS)   // 16 elements per thread, contiguous
#define ROWS_PB   8
#define NBIN_HI   2048               // 11-bit radix passes
#define NBIN_LO   1024               // final 10-bit pass

typedef float vfloat4 __attribute__((ext_vector_type(4)));  // native vec for NT stores

__device__ __forceinline__ uint32_t f2sort(uint32_t b) {
  // monotone map: float total order -> unsigned order (-0 < +0 preserved)
  return (b & 0x80000000u) ? ~b : (b | 0x80000000u);
}
__device__ __forceinline__ uint32_t sort2f(uint32_t u) {
  return (u & 0x80000000u) ? (u ^ 0x80000000u) : ~u;
}

__global__ __launch_bounds__(THREADS) void radix_drop_kernel(
    const float* __restrict__ feat, const float* __restrict__ prop,
    const float* __restrict__ scale, float* __restrict__ out, int nrows)
{
  __shared__ __align__(16) uint32_t lbuf[2][FEAT];   // 32 KB double buffer
  __shared__ uint32_t hist[NBIN_HI];                 // 8 KB histogram
  __shared__ uint32_t wsum[NWAVES];                  // cross-wave scan scratch
  __shared__ uint32_t bc[2];                         // bin / prefix broadcast

  const int tid  = threadIdx.x;
  const int lane = tid & 31;
  const int wid  = tid >> 5;
  const int row0 = blockIdx.x * ROWS_PB;
  if (row0 >= nrows) return;
  const int rend = (row0 + ROWS_PB < nrows) ? (row0 + ROWS_PB) : nrows;

  const float scale0 = scale[0];
  const uint32_t voff = (uint32_t)(tid * 16);        // per-lane byte offset

  // --- gfx1250 async copy: 4 x b128 per wave covers 16KB row across 8 waves.
  // IOFFSET is added to BOTH the global and the LDS address (ISA 08 §4.4).
  // TH_LOAD_NT: row is consumed exactly once -> don't occupy L2 residency.
  auto issue_copy = [&](int row, int buf) {
    uint32_t lds = (uint32_t)(uintptr_t)(&lbuf[buf][0]) + voff;
    const char* g = (const char*)feat + (size_t)row * (FEAT * 4);
    asm volatile(
        "global_load_async_to_lds_b128 %0, %1, %2 offset:0 th:TH_LOAD_NT\n\t"
        "global_load_async_to_lds_b128 %0, %1, %2 offset:4096 th:TH_LOAD_NT\n\t"
        "global_load_async_to_lds_b128 %0, %1, %2 offset:8192 th:TH_LOAD_NT\n\t"
        "global_load_async_to_lds_b128 %0, %1, %2 offset:12288 th:TH_LOAD_NT"
        :
        : "v"(lds), "v"(voff), "s"(g)
        : "memory");
  };

  // --- block-wide inclusive scan: wave32 shuffle scan + cross-wave combine.
  // Only 2 barriers per scan; no LDS traffic for the in-wave part.
  auto block_incl_scan = [&](uint32_t v) -> uint32_t {
    uint32_t s = v;
    #pragma unroll
    for (int off = 1; off < 32; off <<= 1) {
      uint32_t n = __shfl_up(s, off, 32);
      if (lane >= off) s += n;
    }
    if (lane == 31) wsum[wid] = s;                   // wave total
    __syncthreads();
    uint32_t base = 0;
    #pragma unroll
    for (int j = 0; j < NWAVES; ++j)
      base += (j < wid) ? wsum[j] : 0u;
    __syncthreads();                                 // protect wsum for reuse
    return s + base;
  };

  // --- locate histogram bin containing rank `sel_`; returns {bin, cum_before}
  auto find_bin = [&](int nbins, int sel_) -> uint2 {
    const int chunk = nbins / THREADS;
    const int base  = tid * chunk;
    uint32_t own = 0;
    for (int c = 0; c < chunk; ++c) own += hist[base + c];
    uint32_t incl = block_incl_scan(own);
    uint32_t run  = incl - own;                      // exclusive prefix
    for (int c = 0; c < chunk; ++c) {
      uint32_t h = hist[base + c];
      if (h && (int)run <= sel_ && sel_ < (int)(run + h)) {
        bc[0] = (uint32_t)(base + c);
        bc[1] = run;
      }
      run += h;
    }
    __syncthreads();
    uint2 r; r.x = bc[0]; r.y = bc[1];
    __syncthreads();
    return r;
  };

  issue_copy(row0, 0);

  for (int i = 0; row0 + i < rend; ++i) {
    const int row = row0 + i;
    const int cur = i & 1;
    if (row + 1 < rend) {
      issue_copy(row + 1, cur ^ 1);
      asm volatile("s_wait_asynccnt 0x4" ::: "memory");  // row i done (in-order)
    } else {
      asm volatile("s_wait_asynccnt 0x0" ::: "memory");
    }
    __syncthreads();                                      // cross-wave visibility

    // contiguous 16 elements per thread (index order), 4 x ds_load_b128
    uint32_t u[PER_TH];
    {
      const uint4* lrow = (const uint4*)(&lbuf[cur][tid * PER_TH]);
      #pragma unroll
      for (int q = 0; q < 4; ++q) {
        uint4 w = lrow[q];
        u[4*q+0] = f2sort(w.x);
        u[4*q+1] = f2sort(w.y);
        u[4*q+2] = f2sort(w.z);
        u[4*q+3] = f2sort(w.w);
      }
    }

    const float p  = prop[row];
    const float rs = scale0 / p;
    int k = (int)__builtin_rintf((float)FEAT * p);        // round-half-even
    if (k > FEAT) k = FEAT;

    vfloat4* const orow = (vfloat4*)(out + (size_t)row * FEAT + tid * PER_TH);

    if (k <= 0) {                                         // uniform per block
      const vfloat4 z = {0.f, 0.f, 0.f, 0.f};
      #pragma unroll
      for (int q = 0; q < 4; ++q) __builtin_nontemporal_store(z, &orow[q]);
      __syncthreads();
      continue;
    }

    int sel = k - 1;                                      // rank of threshold

    // ---- pass 1: bits [31:21] (11 bits; wide to spread normal-dist exponents)
    #pragma unroll
    for (int c = 0; c < NBIN_HI / THREADS; ++c) hist[tid + c * THREADS] = 0;
    __syncthreads();
    #pragma unroll
    for (int m = 0; m < PER_TH; ++m) atomicAdd(&hist[u[m] >> 21], 1u);
    __syncthreads();
    uint2 r1 = find_bin(NBIN_HI, sel);
    uint32_t pref = r1.x;
    sel -= (int)r1.y;

    // ---- pass 2: bits [20:10]
    #pragma unroll
    for (int c = 0; c < NBIN_HI / THREADS; ++c) hist[tid + c * THREADS] = 0;
    __syncthreads();
    #pragma unroll
    for (int m = 0; m < PER_TH; ++m)
      if ((u[m] >> 21) == pref) atomicAdd(&hist[(u[m] >> 10) & 0x7FFu], 1u);
    __syncthreads();
    uint2 r2 = find_bin(NBIN_HI, sel);
    pref = (pref << 11) | r2.x;
    sel -= (int)r2.y;

    // ---- pass 3: bits [9:0]
    #pragma unroll
    for (int c = 0; c < NBIN_LO / THREADS; ++c) hist[tid + c * THREADS] = 0;
    __syncthreads();
    #pragma unroll
    for (int m = 0; m < PER_TH; ++m)
      if ((u[m] >> 10) == pref) atomicAdd(&hist[u[m] & 0x3FFu], 1u);
    __syncthreads();
    uint2 r3 = find_bin(NBIN_LO, sel);
    const uint32_t t = (pref << 10) | r3.x;               // exact threshold key
    sel -= (int)r3.y;
    const int rkeep = sel + 1;          // # threshold-equal elems kept (stable)

    // ---- stable tie-break: prefix scan of equal-counts in index order
    uint32_t eqc = 0;
    #pragma unroll
    for (int m = 0; m < PER_TH; ++m) eqc += (u[m] == t) ? 1u : 0u;
    uint32_t eincl = block_incl_scan(eqc);
    int erun = (int)(eincl - eqc);

    float vals[PER_TH];
    #pragma unroll
    for (int m = 0; m < PER_TH; ++m) {
      const uint32_t um = u[m];
      bool keep;
      if (um < t)       keep = true;
      else if (um == t) { keep = (erun < rkeep); ++erun; }
      else              keep = false;
      vals[m] = keep ? __uint_as_float(sort2f(um)) * rs : 0.0f;
    }
    #pragma unroll
    for (int q = 0; q < 4; ++q) {
      vfloat4 v = {vals[4*q], vals[4*q+1], vals[4*q+2], vals[4*q+3]};
      __builtin_nontemporal_store(v, &orow[q]);           // write-once stream
    }

    __syncthreads();   // all reads of lbuf[cur] done before it is refilled
  }
}

extern "C" void kernel_launch(void* const* d_in, const int* in_sizes, int n_in,
                              void* d_out, int out_size, void* d_ws, size_t ws_size,
                              hipStream_t stream) {
  (void)n_in; (void)out_size; (void)d_ws; (void)ws_size;
  const float* feat  = (const float*)d_in[0];
  const float* prop  = (const float*)d_in[1];
  const float* scale = (const float*)d_in[2];
  float* out = (float*)d_out;
  const int nrows  = in_sizes[1];                       // BATCH (prop length)
  const int blocks = (nrows + ROWS_PB - 1) / ROWS_PB;
  radix_drop_kernel<<<blocks, THREADS, 0, stream>>>(feat, prop, scale, out, nrows);
}